// SoftVoronoi_14078902797091
// MI455X (gfx1250) — compile-verified
//
#include <hip/hip_runtime.h>
#include <math.h>

typedef __attribute__((ext_vector_type(2))) float v2f;
typedef __attribute__((ext_vector_type(4))) float v4f;
typedef __attribute__((ext_vector_type(8))) float v8f;

#define LOG2E 1.44269504088896340736f

// One 16-query x 4-neuron chunk: k values (B fragment) + coefficient A fragment
// (row0 = activities, row1 = ones, rows 2..15 = 0).
__device__ __forceinline__ void sv_chunk(const float* __restrict__ s_pos,
                                         const float* __restrict__ s_act,
                                         int nb, int halfsel, int row,
                                         float qx, float qy,
                                         float bl2, float ebias, float invR2,
                                         v2f& amat, v2f& bmat)
{
    const int pidx = (nb >> 1) + halfsel;            // neuron pair for this lane
    const v4f pp = ((const v4f*)s_pos)[pidx];        // px0,py0,px1,py1
    const v2f aa = ((const v2f*)s_act)[pidx];        // act0, act1

    // pair (q, n0)
    float dx0 = qx - pp.x, dy0 = qy - pp.y;
    float r20 = __builtin_fmaf(dx0, dx0, __builtin_fmaf(dy0, dy0, 1e-8f));
    float r0  = __builtin_amdgcn_sqrtf(r20);
    float t0  = fmaxf(__builtin_fmaf(-r20, invR2, 1.0f), 0.0f); // 1-(r/R)^2 clipped
    float t02 = t0 * t0, t04 = t02 * t02, t08 = t04 * t04;
    float e0  = __builtin_amdgcn_exp2f(__builtin_fmaf(bl2, r0, -ebias)); // exp(-b(R-r))
    float m0  = __builtin_amdgcn_rcpf(1.0f + e0);    // sigmoid(b(R-r))
    float k0  = (t08 * t02) * m0;                    // bump^10 * mask

    // pair (q, n1)
    float dx1 = qx - pp.z, dy1 = qy - pp.w;
    float r21 = __builtin_fmaf(dx1, dx1, __builtin_fmaf(dy1, dy1, 1e-8f));
    float r1  = __builtin_amdgcn_sqrtf(r21);
    float t1  = fmaxf(__builtin_fmaf(-r21, invR2, 1.0f), 0.0f);
    float t12 = t1 * t1, t14 = t12 * t12, t18 = t14 * t14;
    float e1  = __builtin_amdgcn_exp2f(__builtin_fmaf(bl2, r1, -ebias));
    float m1  = __builtin_amdgcn_rcpf(1.0f + e1);
    float k1  = (t18 * t12) * m1;

    bmat.x = k0;
    bmat.y = k1;
    amat.x = (row == 0) ? aa.x : ((row == 1) ? 1.0f : 0.0f);
    amat.y = (row == 0) ? aa.y : ((row == 1) ? 1.0f : 0.0f);
}

__global__ __launch_bounds__(256) void SoftVoronoi_wmma_kernel(
    const float* __restrict__ positions,   // (N,2)
    const float* __restrict__ activities,  // (N,)
    const float* __restrict__ query,       // (M,2)
    const float* __restrict__ p_ax, const float* __restrict__ p_ay,
    const float* __restrict__ p_tx, const float* __restrict__ p_ty,
    const float* __restrict__ p_sigma, const float* __restrict__ p_beta,
    float* __restrict__ out, int N, int M)
{
    extern __shared__ float smem[];
    float* s_pos = smem;           // pair-packed [px0,py0,px1,py1] : 2N floats
    float* s_act = smem + 2 * N;   // N floats

    const float ax = p_ax[0], ay = p_ay[0], tx = p_tx[0], ty = p_ty[0];
    const float sigma = p_sigma[0], beta = p_beta[0];
    const float R     = logf(1.0f + expf(sigma)) + 1e-6f;  // softplus(sigma)+1e-6
    const float bsp   = logf(1.0f + expf(beta)) + 1e-6f;   // softplus(beta)+1e-6
    const float invR2 = 1.0f / (R * R);
    const float bl2   = bsp * LOG2E;     // sigmoid exponent in log2 domain
    const float ebias = bl2 * R;         // e = exp2(bl2*r - bl2*R)

    const int tid = threadIdx.x;
    // affine-transform neuron positions once per block into LDS
    for (int i = tid; i < N; i += blockDim.x) {
        float px = ax * (positions[2 * i + 0] - 0.5f) + tx + 0.5f;
        float py = ay * (positions[2 * i + 1] - 0.5f) + ty + 0.5f;
        int p = i >> 1, s = i & 1;
        s_pos[p * 4 + s * 2 + 0] = px;
        s_pos[p * 4 + s * 2 + 1] = py;
        s_act[i] = activities[i];
    }
    __syncthreads();

    const int lane    = tid & 31;
    const int wave    = tid >> 5;
    const int row     = lane & 15;     // A-matrix row / query column / output query
    const int halfsel = lane >> 4;     // 0 -> K=0,1 ; 1 -> K=2,3
    const int qbase   = (blockIdx.x * (blockDim.x >> 5) + wave) * 16;
    const int q       = qbase + row;
    const int qload   = (q < M) ? q : (M - 1);  // clamp so EXEC stays full for WMMA
    const float qx = query[2 * qload + 0];
    const float qy = query[2 * qload + 1];

    v8f acc0 = {};
    v8f acc1 = {};

    int nb = 0;
    // 16 neurons (4 WMMA chunks) per iteration, 2 alternating accumulators
    for (; nb + 16 <= N; nb += 16) {
        v2f a0, b0, a1, b1, a2, b2, a3, b3;
        sv_chunk(s_pos, s_act, nb +  0, halfsel, row, qx, qy, bl2, ebias, invR2, a0, b0);
        sv_chunk(s_pos, s_act, nb +  4, halfsel, row, qx, qy, bl2, ebias, invR2, a1, b1);
        sv_chunk(s_pos, s_act, nb +  8, halfsel, row, qx, qy, bl2, ebias, invR2, a2, b2);
        sv_chunk(s_pos, s_act, nb + 12, halfsel, row, qx, qy, bl2, ebias, invR2, a3, b3);
        acc0 = __builtin_amdgcn_wmma_f32_16x16x4_f32(false, a0, false, b0,
                                                     (short)0, acc0, false, false);
        acc1 = __builtin_amdgcn_wmma_f32_16x16x4_f32(false, a1, false, b1,
                                                     (short)0, acc1, false, false);
        acc0 = __builtin_amdgcn_wmma_f32_16x16x4_f32(false, a2, false, b2,
                                                     (short)0, acc0, false, false);
        acc1 = __builtin_amdgcn_wmma_f32_16x16x4_f32(false, a3, false, b3,
                                                     (short)0, acc1, false, false);
    }
    for (; nb + 4 <= N; nb += 4) {  // tail (N % 16 != 0, in multiples of 4)
        v2f a0, b0;
        sv_chunk(s_pos, s_act, nb, halfsel, row, qx, qy, bl2, ebias, invR2, a0, b0);
        acc0 = __builtin_amdgcn_wmma_f32_16x16x4_f32(false, a0, false, b0,
                                                     (short)0, acc0, false, false);
    }

    // D[0][col] = num, D[1][col] = den, col = lane (queries) for lanes 0..15
    if (lane < 16 && q < M) {
        float num = acc0[0] + acc1[0];
        float den = acc0[1] + acc1[1] + 1e-8f;
        out[q] = num / den;
    }
}

extern "C" void kernel_launch(void* const* d_in, const int* in_sizes, int n_in,
                              void* d_out, int out_size, void* d_ws, size_t ws_size,
                              hipStream_t stream) {
    const float* positions  = (const float*)d_in[0];
    const float* activities = (const float*)d_in[1];
    const float* query      = (const float*)d_in[2];
    const float* p_ax    = (const float*)d_in[3];
    const float* p_ay    = (const float*)d_in[4];
    const float* p_tx    = (const float*)d_in[5];
    const float* p_ty    = (const float*)d_in[6];
    const float* p_sigma = (const float*)d_in[7];
    const float* p_beta  = (const float*)d_in[8];
    float* out = (float*)d_out;

    const int N = in_sizes[1];          // 1024 neurons
    const int M = in_sizes[2] / 2;      // 262144 query points

    const int block = 256;                       // 8 waves -> 128 queries/block
    const int qpb = (block / 32) * 16;
    const int grid = (M + qpb - 1) / qpb;
    const size_t shmem = (size_t)3 * N * sizeof(float);

    SoftVoronoi_wmma_kernel<<<grid, block, shmem, stream>>>(
        positions, activities, query,
        p_ax, p_ay, p_tx, p_ty, p_sigma, p_beta,
        out, N, M);
}